// PathFinderNet_37512244364092
// MI455X (gfx1250) — compile-verified
//
#include <hip/hip_runtime.h>
#include <math.h>

#define B_   64
#define S_   256
#define NB_  36
#define NE_  120
#define EPSF 1e-6f

typedef float v2f __attribute__((ext_vector_type(2)));
typedef float v8f __attribute__((ext_vector_type(8)));

// ---------------------------------------------------------------- conv1: (B,1,32,32) -> (B,32,28,28)
__global__ __launch_bounds__(128) void k_conv1(const float* __restrict__ img,
                                               const float* __restrict__ w,
                                               const float* __restrict__ bias,
                                               float* __restrict__ out) {
  const int oc = blockIdx.x, b = blockIdx.y, tid = threadIdx.x;
  __shared__ float sIn[32 * 32];
  for (int i = tid; i < 1024; i += 128) sIn[i] = img[b * 1024 + i];
  float wr[25];
#pragma unroll
  for (int i = 0; i < 25; ++i) wr[i] = w[oc * 25 + i];
  const float bb = bias[oc];
  __syncthreads();
  for (int p = tid; p < 784; p += 128) {
    const int y = p / 28, x = p % 28;
    float acc = bb;
#pragma unroll
    for (int ky = 0; ky < 5; ++ky)
#pragma unroll
      for (int kx = 0; kx < 5; ++kx)
        acc += sIn[(y + ky) * 32 + x + kx] * wr[ky * 5 + kx];
    out[((b * 32 + oc) * 28 + y) * 28 + x] = acc;
  }
}

// ------------------------------------------- conv2 + 2x2 maxpool: (B,32,28,28) -> (B,32,12,12)
__global__ __launch_bounds__(128) void k_conv2pool(const float* __restrict__ in,
                                                   const float* __restrict__ w,
                                                   const float* __restrict__ bias,
                                                   float* __restrict__ out) {
  const int oc = blockIdx.x, b = blockIdx.y, tid = threadIdx.x;
  __shared__ float sw[800];
  for (int i = tid; i < 800; i += 128) sw[i] = w[oc * 800 + i];
  __syncthreads();
  const float bb = bias[oc];
  for (int p = tid; p < 144; p += 128) {
    const int py = p / 12, px = p % 12;
    const int y0 = py * 2, x0 = px * 2;
    float a00 = 0.f, a01 = 0.f, a10 = 0.f, a11 = 0.f;
    for (int ic = 0; ic < 32; ++ic) {
      const float* ip = in + (b * 32 + ic) * 784;
      const float* wp = sw + ic * 25;
#pragma unroll
      for (int ky = 0; ky < 5; ++ky) {
        const float* r0 = ip + (y0 + ky) * 28 + x0;
        const float* r1 = r0 + 28;
#pragma unroll
        for (int kx = 0; kx < 5; ++kx) {
          const float ww = wp[ky * 5 + kx];
          a00 += ww * r0[kx];
          a01 += ww * r0[kx + 1];
          a10 += ww * r1[kx];
          a11 += ww * r1[kx + 1];
        }
      }
    }
    out[((b * 32 + oc) * 12 + py) * 12 + px] =
        fmaxf(fmaxf(a00, a01), fmaxf(a10, a11)) + bb;
  }
}

// ---------------------------------------------------------------- conv3: (B,32,12,12) -> (B,64,8,8)
__global__ __launch_bounds__(64) void k_conv3(const float* __restrict__ in,
                                              const float* __restrict__ w,
                                              const float* __restrict__ bias,
                                              float* __restrict__ out) {
  const int oc = blockIdx.x, b = blockIdx.y, tid = threadIdx.x;
  __shared__ float sIn[32 * 144];  // 18KB
  __shared__ float sw[800];
  for (int i = tid; i < 4608; i += 64) sIn[i] = in[b * 4608 + i];
  for (int i = tid; i < 800; i += 64) sw[i] = w[oc * 800 + i];
  __syncthreads();
  const int y = tid / 8, x = tid % 8;
  float acc = bias[oc];
  for (int ic = 0; ic < 32; ++ic) {
    const float* ip = sIn + ic * 144;
    const float* wp = sw + ic * 25;
#pragma unroll
    for (int ky = 0; ky < 5; ++ky)
#pragma unroll
      for (int kx = 0; kx < 5; ++kx)
        acc += ip[(y + ky) * 12 + x + kx] * wp[ky * 5 + kx];
  }
  out[((b * 64 + oc) * 8 + y) * 8 + x] = acc;
}

// -------------------------------- conv4 + 2x2 maxpool + flatten: (B,64,8,8) -> emb (B,256)
__global__ __launch_bounds__(256) void k_conv4pool(const float* __restrict__ in,
                                                   const float* __restrict__ w,
                                                   const float* __restrict__ bias,
                                                   float* __restrict__ emb) {
  const int b = blockIdx.x, tid = threadIdx.x;
  const int oc = tid >> 2, pp = tid & 3;
  const int py = pp >> 1, px = pp & 1;
  __shared__ float sIn[64 * 64];  // 16KB
  for (int i = tid; i < 4096; i += 256) sIn[i] = in[b * 4096 + i];
  __syncthreads();
  const int y0 = py * 2, x0 = px * 2;
  float a00 = 0.f, a01 = 0.f, a10 = 0.f, a11 = 0.f;
  for (int ic = 0; ic < 64; ++ic) {
    const float* ip = sIn + ic * 64;
    const float* wp = w + (oc * 64 + ic) * 25;
#pragma unroll
    for (int ky = 0; ky < 5; ++ky) {
      const float* r0 = ip + (y0 + ky) * 8 + x0;
      const float* r1 = r0 + 8;
#pragma unroll
      for (int kx = 0; kx < 5; ++kx) {
        const float ww = wp[ky * 5 + kx];
        a00 += ww * r0[kx];
        a01 += ww * r0[kx + 1];
        a10 += ww * r1[kx];
        a11 += ww * r1[kx + 1];
      }
    }
  }
  emb[b * 256 + oc * 4 + py * 2 + px] =
      fmaxf(fmaxf(a00, a01), fmaxf(a10, a11)) + bias[oc];
}

// ------------------------- heads: sigmoid(relu(emb@W1+b1)@W2+b2) via V_WMMA_F32_16X16X4_F32
__global__ __launch_bounds__(256) void k_heads(const float* __restrict__ emb,
                                               const float* __restrict__ wc1, const float* __restrict__ bc1,
                                               const float* __restrict__ wc2, const float* __restrict__ bc2,
                                               const float* __restrict__ we1, const float* __restrict__ be1,
                                               const float* __restrict__ we2, const float* __restrict__ be2,
                                               float* __restrict__ p_conn, float* __restrict__ p_ep) {
  const int head = blockIdx.x;
  const float* W1 = head ? we1 : wc1;
  const float* B1 = head ? be1 : bc1;
  const float* W2 = head ? we2 : wc2;
  const float* B2 = head ? be2 : bc2;
  float* OUT = head ? p_ep : p_conn;
  const int NOUT = head ? NB_ : NE_;
  const int NT2 = head ? 3 : 8;  // ceil(NOUT/16)

  __shared__ float Hs[64 * 256];  // 64KB: H = relu(emb@W1+b1)
  const int tid = threadIdx.x;
  const int wv = tid >> 5, lane = tid & 31;
  const int half = lane >> 4, l = lane & 15;

  // ---- phase 1: 64 x 256 = 4x16 tiles of 16x16, 8 waves x 8 tiles ----
  for (int i = 0; i < 8; ++i) {
    const int t = wv * 8 + i;
    const int mt = t >> 4, nt = t & 15;
    v8f c = {0.f, 0.f, 0.f, 0.f, 0.f, 0.f, 0.f, 0.f};
    const float* arow = emb + (mt * 16 + l) * 256;
    const float* bcol = W1 + nt * 16 + l;
    for (int k = 0; k < 256; k += 4) {
      const int ka = k + 2 * half;  // lanes 0-15: K=k,k+1 ; lanes 16-31: K=k+2,k+3
      v2f a, bv;
      a.x = arow[ka];
      a.y = arow[ka + 1];
      bv.x = bcol[ka * 256];
      bv.y = bcol[(ka + 1) * 256];
      c = __builtin_amdgcn_wmma_f32_16x16x4_f32(false, a, false, bv, (short)0, c,
                                                false, false);
    }
    const int n = nt * 16 + l;
    const float bb = B1[n];
#pragma unroll
    for (int r = 0; r < 8; ++r) {
      const int m = mt * 16 + half * 8 + r;
      const float v = c[r] + bb;
      Hs[m * 256 + n] = v > 0.f ? v : 0.f;
    }
  }
  __syncthreads();

  // ---- phase 2: 64 x NOUT (N padded to 16) ----
  const int tiles2 = 4 * NT2;
  for (int t = wv; t < tiles2; t += 8) {
    const int mt = t / NT2, nt = t % NT2;
    const int n = nt * 16 + l;
    const int nc = n < NOUT ? n : NOUT - 1;      // clamp (no EXEC divergence
    const float mask = n < NOUT ? 1.f : 0.f;     //  around the WMMA)
    v8f c = {0.f, 0.f, 0.f, 0.f, 0.f, 0.f, 0.f, 0.f};
    const float* arow = Hs + (mt * 16 + l) * 256;
    for (int k = 0; k < 256; k += 4) {
      const int ka = k + 2 * half;
      v2f a, bv;
      a.x = arow[ka];
      a.y = arow[ka + 1];
      bv.x = W2[ka * NOUT + nc] * mask;
      bv.y = W2[(ka + 1) * NOUT + nc] * mask;
      c = __builtin_amdgcn_wmma_f32_16x16x4_f32(false, a, false, bv, (short)0, c,
                                                false, false);
    }
    if (n < NOUT) {
      const float bb = B2[n];
#pragma unroll
      for (int r = 0; r < 8; ++r) {
        const int m = mt * 16 + half * 8 + r;
        const float z = c[r] + bb;
        float p = 1.f / (1.f + expf(-z));
        p = fminf(fmaxf(p, EPSF), 1.f - EPSF);
        OUT[m * NOUT + n] = p;  // stored already clipped (matches jnp.clip usage)
      }
    }
  }
}

// -------------------- per-batch: logcat = log(pe/sum(pe)), log(pc), log1p(-pc)
__global__ __launch_bounds__(128) void k_logs(const float* __restrict__ p_conn,
                                              const float* __restrict__ p_ep,
                                              float* __restrict__ logcat,
                                              float* __restrict__ logpc,
                                              float* __restrict__ log1mpc) {
  const int b = blockIdx.x, tid = threadIdx.x;
  __shared__ float spe[NB_];
  __shared__ float slogsum;
  if (tid < NB_) spe[tid] = p_ep[b * NB_ + tid];
  __syncthreads();
  if (tid == 0) {
    float s = 0.f;
    for (int j = 0; j < NB_; ++j) s += spe[j];
    slogsum = logf(s);
  }
  __syncthreads();
  if (tid < NB_) logcat[b * NB_ + tid] = logf(spe[tid]) - slogsum;
  for (int e = tid; e < NE_; e += 128) {
    const float p = p_conn[b * NE_ + e];
    logpc[b * NE_ + e] = logf(p);
    log1mpc[b * NE_ + e] = log1pf(-p);
  }
}

// ---- per sample: Bernoulli edges, Gumbel argmax s/d, bitset-BFS reachability, log-weight
__global__ __launch_bounds__(128) void k_sample(const float* __restrict__ edge_noise,
                                                const float* __restrict__ u_s,
                                                const float* __restrict__ u_d,
                                                const float* __restrict__ p_conn,
                                                const float* __restrict__ logpc,
                                                const float* __restrict__ log1mpc,
                                                const float* __restrict__ logcat,
                                                const int* __restrict__ edges,
                                                float* __restrict__ logw,
                                                float* __restrict__ exv) {
  const int tid = threadIdx.x;
  const int id = blockIdx.x * 128 + tid;  // one block spans one b (128 | 256)
  const int b = id >> 8;

  __shared__ unsigned long long sAdj[128 * NB_];  // 36KB: per-lane adjacency masks
  __shared__ float s_pc[NE_], s_lpc[NE_], s_l1m[NE_], s_lcat[NB_];
  __shared__ int s_eu[NE_], s_ev[NE_];
  for (int i = tid; i < NE_; i += 128) {
    s_pc[i] = p_conn[b * NE_ + i];
    s_lpc[i] = logpc[b * NE_ + i];
    s_l1m[i] = log1mpc[b * NE_ + i];
    s_eu[i] = edges[2 * i];
    s_ev[i] = edges[2 * i + 1];
  }
  if (tid < NB_) s_lcat[tid] = logcat[b * NB_ + tid];
  __syncthreads();

  unsigned long long* adj = sAdj + tid * NB_;
#pragma unroll
  for (int j = 0; j < NB_; ++j) adj[j] = 0ull;

  float lw = 0.f;
  const float* en = edge_noise + (size_t)id * NE_;
  for (int e = 0; e < NE_; ++e) {
    const bool cn = en[e] < s_pc[e];
    lw += cn ? s_lpc[e] : s_l1m[e];
    if (cn) {
      const int u = s_eu[e], v = s_ev[e];
      adj[u] |= (1ull << v);
      adj[v] |= (1ull << u);
    }
  }

  // Gumbel-max categorical sampling (first max wins, matching jnp.argmax)
  const float* us = u_s + (size_t)id * NB_;
  const float* ud = u_d + (size_t)id * NB_;
  float bs = -1e30f, bd = -1e30f;
  int si = 0, di = 0;
  for (int j = 0; j < NB_; ++j) {
    float uu = fminf(fmaxf(us[j], EPSF), 1.f - EPSF);
    float v = s_lcat[j] - logf(-logf(uu));
    if (v > bs) { bs = v; si = j; }
    uu = fminf(fmaxf(ud[j], EPSF), 1.f - EPSF);
    v = s_lcat[j] - logf(-logf(uu));
    if (v > bd) { bd = v; di = j; }
  }

  // bitset BFS: reach = nodes reachable from si via paths of length >= 1
  unsigned long long reach = adj[si];
  unsigned long long frontier = reach;
  while (frontier) {
    unsigned long long next = 0ull, f = frontier;
    while (f) {
      const int u = __builtin_ctzll(f);
      f &= f - 1;
      next |= adj[u];
    }
    frontier = next & ~reach;
    reach |= next;
  }

  exv[id] = (float)((reach >> di) & 1ull);
  logw[id] = lw + s_lcat[si] + s_lcat[di];
}

// -------------------------- per batch: softmax over S, p1 = sum(w*exists), out (B,2)
__global__ __launch_bounds__(256) void k_final(const float* __restrict__ logw,
                                               const float* __restrict__ exv,
                                               float* __restrict__ out) {
  const int b = blockIdx.x, tid = threadIdx.x;
  __shared__ float sm[S_];
  const float lw = logw[b * S_ + tid];
  const float ex = exv[b * S_ + tid];

  sm[tid] = lw;
  __syncthreads();
  for (int off = 128; off; off >>= 1) {
    if (tid < off) sm[tid] = fmaxf(sm[tid], sm[tid + off]);
    __syncthreads();
  }
  const float mx = sm[0];
  __syncthreads();

  const float w = expf(lw - mx);
  sm[tid] = w;
  __syncthreads();
  for (int off = 128; off; off >>= 1) {
    if (tid < off) sm[tid] += sm[tid + off];
    __syncthreads();
  }
  const float wsum = sm[0];
  __syncthreads();

  sm[tid] = w * ex;
  __syncthreads();
  for (int off = 128; off; off >>= 1) {
    if (tid < off) sm[tid] += sm[tid + off];
    __syncthreads();
  }
  if (tid == 0) {
    const float p1 = sm[0] / wsum;
    out[b * 2 + 0] = 1.f - p1;
    out[b * 2 + 1] = p1;
  }
}

extern "C" void kernel_launch(void* const* d_in, const int* in_sizes, int n_in,
                              void* d_out, int out_size, void* d_ws, size_t ws_size,
                              hipStream_t stream) {
  const float* image = (const float*)d_in[0];
  const float* edge_noise = (const float*)d_in[1];
  const float* u_s = (const float*)d_in[2];
  const float* u_d = (const float*)d_in[3];
  const float* cw1 = (const float*)d_in[4];
  const float* cb1 = (const float*)d_in[5];
  const float* cw2 = (const float*)d_in[6];
  const float* cb2 = (const float*)d_in[7];
  const float* cw3 = (const float*)d_in[8];
  const float* cb3 = (const float*)d_in[9];
  const float* cw4 = (const float*)d_in[10];
  const float* cb4 = (const float*)d_in[11];
  const float* wc1 = (const float*)d_in[12];
  const float* bc1 = (const float*)d_in[13];
  const float* wc2 = (const float*)d_in[14];
  const float* bc2 = (const float*)d_in[15];
  const float* we1 = (const float*)d_in[16];
  const float* be1 = (const float*)d_in[17];
  const float* we2 = (const float*)d_in[18];
  const float* be2 = (const float*)d_in[19];
  const int* edges = (const int*)d_in[20];
  float* out = (float*)d_out;

  float* ws = (float*)d_ws;
  float* h1 = ws;                          // (64,32,28,28) = 1605632
  float* h2 = h1 + 1605632;                // (64,32,12,12) = 294912
  float* h3 = h2 + 294912;                 // (64,64,8,8)   = 262144
  float* emb = h3 + 262144;                // (64,256)      = 16384
  float* p_conn = emb + 16384;             // (64,120)
  float* p_ep = p_conn + B_ * NE_;         // (64,36)
  float* logcat = p_ep + B_ * NB_;         // (64,36)
  float* logpc = logcat + B_ * NB_;        // (64,120)
  float* log1mpc = logpc + B_ * NE_;       // (64,120)
  float* logw = log1mpc + B_ * NE_;        // (64,256)
  float* exv = logw + B_ * S_;             // (64,256)

  k_conv1<<<dim3(32, B_), 128, 0, stream>>>(image, cw1, cb1, h1);
  k_conv2pool<<<dim3(32, B_), 128, 0, stream>>>(h1, cw2, cb2, h2);
  k_conv3<<<dim3(64, B_), 64, 0, stream>>>(h2, cw3, cb3, h3);
  k_conv4pool<<<B_, 256, 0, stream>>>(h3, cw4, cb4, emb);
  k_heads<<<2, 256, 0, stream>>>(emb, wc1, bc1, wc2, bc2, we1, be1, we2, be2,
                                 p_conn, p_ep);
  k_logs<<<B_, 128, 0, stream>>>(p_conn, p_ep, logcat, logpc, log1mpc);
  k_sample<<<(B_ * S_) / 128, 128, 0, stream>>>(edge_noise, u_s, u_d, p_conn,
                                                logpc, log1mpc, logcat, edges,
                                                logw, exv);
  k_final<<<B_, 256, 0, stream>>>(logw, exv, out);
}